// torchMetric_62912680951985
// MI455X (gfx1250) — compile-verified
//
#include <hip/hip_runtime.h>

typedef __attribute__((ext_vector_type(2))) float v2f;
typedef __attribute__((ext_vector_type(8))) float v8f;

#define C1F 1.0e-4f    // 0.01^2
#define C2F 9.0e-4f    // 0.03^2

// ---- workspace layout (bytes) ----
// [0..31]   double mse[4]
// [32..63]  double ssim[4]
// [64..79]  unsigned mm[4] : {max_p, min_p, max_g, min_g} (order-mapped uints)
// [80..95]  float scale[4] : {a_p, b_p, a_g, b_g}

__device__ __forceinline__ unsigned mapf(float f) {
  unsigned u = __float_as_uint(f);
  return (u & 0x80000000u) ? ~u : (u | 0x80000000u);
}
__device__ __forceinline__ float unmapf(unsigned m) {
  unsigned u = (m & 0x80000000u) ? (m ^ 0x80000000u) : ~m;
  return __uint_as_float(u);
}

__global__ void k_init(void* wsp) {
  int t = threadIdx.x;
  double* d = (double*)wsp;
  if (t < 8) d[t] = 0.0;
  unsigned* mm = (unsigned*)((char*)wsp + 64);
  if (t < 4) mm[t] = (t & 1) ? 0xFFFFFFFFu : 0u;  // even=max slot, odd=min slot
}

__global__ __launch_bounds__(256) void k_minmax(const float* __restrict__ x,
                                                unsigned long long nElem,
                                                unsigned* mm, int slotBase) {
  __shared__ unsigned smx[256], smn[256];
  const unsigned tid = threadIdx.x;
  unsigned long long i = (unsigned long long)blockIdx.x * blockDim.x + tid;
  const unsigned long long stride = (unsigned long long)gridDim.x * blockDim.x;
  unsigned vmax = 0u, vmin = 0xFFFFFFFFu;
  for (; i < nElem; i += stride) {
    __builtin_prefetch(x + i + 4 * stride, 0, 1);  // global_prefetch_b8
    unsigned m = mapf(x[i]);
    vmax = vmax > m ? vmax : m;
    vmin = vmin < m ? vmin : m;
  }
  smx[tid] = vmax; smn[tid] = vmin;
  __syncthreads();
  for (int s = 128; s > 0; s >>= 1) {
    if (tid < (unsigned)s) {
      smx[tid] = smx[tid] > smx[tid + s] ? smx[tid] : smx[tid + s];
      smn[tid] = smn[tid] < smn[tid + s] ? smn[tid] : smn[tid + s];
    }
    __syncthreads();
  }
  if (tid == 0) {
    atomicMax(&mm[slotBase + 0], smx[0]);
    atomicMin(&mm[slotBase + 1], smn[0]);
  }
}

__global__ void k_scale(void* wsp) {
  if (threadIdx.x != 0 || blockIdx.x != 0) return;
  const unsigned* mm = (const unsigned*)((const char*)wsp + 64);
  float* scl = (float*)((char*)wsp + 80);
  for (int a = 0; a < 2; ++a) {
    float mx = unmapf(mm[2 * a + 0]);
    float mn = unmapf(mm[2 * a + 1]);
    bool f1 = mx > 128.0f;
    float m1 = f1 ? mn * (1.0f / 255.0f) : mn;
    bool f2 = m1 < -0.5f;
    float A = (f1 ? (1.0f / 255.0f) : 1.0f) * (f2 ? 0.5f : 1.0f);
    float B = f2 ? 0.5f : 0.0f;
    scl[2 * a + 0] = A;
    scl[2 * a + 1] = B;
  }
}

__device__ __forceinline__ void async_ld32(unsigned lds_off, unsigned long long gaddr) {
  asm volatile("global_load_async_to_lds_b32 %0, %1, off"
               :: "v"(lds_off), "v"(gaddr) : "memory");
}

__global__ __launch_bounds__(256) void k_main(const float* __restrict__ pred,
                                              const float* __restrict__ gt,
                                              void* wsp) {
  // Large static LDS (~138.5KB of the WGP's 320KB) -> 2 workgroups/WGP.
  __shared__ float sp[16][26][26];   // raw pred tile + halo
  __shared__ float sg[16][26][26];   // raw gt tile + halo
  __shared__ float t1[16][32][16];   // W-blurred field (rows 26..31 = dead pad)
  __shared__ float t2[16][256];      // H-blurred field, [depth][s = h*16+w]
  __shared__ float g16[16][16];      // depth-blur matrix (replicate pad folded)
  __shared__ float wbm[32][16];      // W-blur band matrix (K x N), zero-padded K
  __shared__ float hbm[16][32];      // H-blur band matrix (M x K), zero-padded K
  __shared__ float red[256];

  const int tid = threadIdx.x;
  const int n  = blockIdx.z;
  const int h0 = blockIdx.y * 16;
  const int w0 = blockIdx.x * 16;

  // Gaussian taps: double precision as the reference, cast to f32.
  float gw[11];
  {
    double t0[11]; double s = 0.0;
#pragma unroll
    for (int t = 0; t < 11; ++t) { double x = (double)(t - 5); t0[t] = exp(-(x * x) / 4.5); s += t0[t]; }
#pragma unroll
    for (int t = 0; t < 11; ++t) gw[t] = (float)(t0[t] / s);
  }

  // Build the three blur matrices in LDS.
  {
    int d = tid >> 4, k = tid & 15;
    float acc = 0.f;
#pragma unroll
    for (int t = 0; t < 11; ++t) {
      int kk = d + t - 5;
      kk = kk < 0 ? 0 : (kk > 15 ? 15 : kk);
      if (kk == k) acc += gw[t];
    }
    g16[d][k] = acc;
  }
  for (int e = tid; e < 512; e += 256) {        // wbm[k][w] = g[k-w]
    int k = e >> 4, w = e & 15;
    int t = k - w;
    wbm[k][w] = (t >= 0 && t <= 10) ? gw[t] : 0.f;
  }
  for (int e = tid; e < 512; e += 256) {        // hbm[h][k] = g[k-h]
    int h = e >> 5, k = e & 31;
    int t = k - h;
    hbm[h][k] = (t >= 0 && t <= 10) ? gw[t] : 0.f;
  }

  const float* scl = (const float*)((const char*)wsp + 80);
  const float ap = scl[0], bp = scl[1], ag = scl[2], bg = scl[3];

  // Async-stage RAW p,g tile (clamped halo = replicate padding). Rerange is
  // affine and blurs are linear, so it is applied analytically after blurring.
  const size_t baseN = (size_t)n * 16 * 512 * 512;
  for (int e = tid; e < 16 * 26 * 26; e += 256) {
    int d  = e / 676;
    int r  = e % 676;
    int hh = r / 26, ww = r % 26;
    int h = h0 - 5 + hh; h = h < 0 ? 0 : (h > 511 ? 511 : h);
    int w = w0 - 5 + ww; w = w < 0 ? 0 : (w > 511 ? 511 : w);
    size_t gi = baseN + ((size_t)d * 512 + h) * 512 + w;
    async_ld32((unsigned)(uintptr_t)&sp[d][hh][ww], (unsigned long long)(uintptr_t)(pred + gi));
    async_ld32((unsigned)(uintptr_t)&sg[d][hh][ww], (unsigned long long)(uintptr_t)(gt + gi));
  }
  asm volatile("s_wait_asynccnt 0" ::: "memory");
  __syncthreads();

  const int wv = tid >> 5, ln = tid & 31;
  const int half = ln >> 4, lr = ln & 15;
  const int koff = 2 * half;

  // Hoist all constant-matrix WMMA fragments into registers (loaded once).
  v2f wbf[8], hbf[8], gf[4];
#pragma unroll
  for (int s8 = 0; s8 < 8; ++s8) {
    int c0 = s8 * 4 + koff;
    wbf[s8][0] = wbm[c0][lr];     wbf[s8][1] = wbm[c0 + 1][lr];
    hbf[s8][0] = hbm[lr][c0];     hbf[s8][1] = hbm[lr][c0 + 1];
  }
#pragma unroll
  for (int s4 = 0; s4 < 4; ++s4) {
    int c0 = s4 * 4 + koff;
    gf[s4][0] = g16[lr][c0];      gf[s4][1] = g16[lr][c0 + 1];
  }

  // MSE partial (reranged) over this tile's exact-partition central region.
  float msum = 0.f;
  for (int o = tid; o < 4096; o += 256) {
    int d = o >> 8, h = (o >> 4) & 15, w = o & 15;
    float df = fmaf(sp[d][h + 5][w + 5], ap, bp) - fmaf(sg[d][h + 5][w + 5], ag, bg);
    msum = fmaf(df, df, msum);
  }

  v8f blur[5][2];
#pragma unroll
  for (int f = 0; f < 5; ++f) { blur[f][0] = {}; blur[f][1] = {}; }

#pragma unroll
  for (int f = 0; f < 5; ++f) {
    // ---- W blur as WMMA: OUT[h][w'] = sum_k F[h][k] * wbm[k][w'].
    // 32 GEMM groups (16 depths x 2 row-batches of 16 covering 26 halo rows).
#pragma unroll
    for (int i = 0; i < 4; ++i) {
      int grp = wv + 8 * i;
      int d = grp >> 1;
      int rowbase = (grp & 1) * 16;
      int hrow = rowbase + lr; hrow = hrow > 25 ? 25 : hrow;
      v8f acc = {};
#pragma unroll
      for (int kk = 0; kk < 32; kk += 4) {
        int c0 = kk + koff, c1 = c0 + 1;
        int cc0 = c0 > 25 ? 25 : c0, cc1 = c1 > 25 ? 25 : c1;  // wbm rows >=26 are 0
        float p0 = sp[d][hrow][cc0], q0 = sg[d][hrow][cc0];
        float p1 = sp[d][hrow][cc1], q1 = sg[d][hrow][cc1];
        v2f a;
        a[0] = (f == 0) ? p0 : (f == 1) ? q0 : (f == 2) ? p0 * p0 : (f == 3) ? q0 * q0 : p0 * q0;
        a[1] = (f == 0) ? p1 : (f == 1) ? q1 : (f == 2) ? p1 * p1 : (f == 3) ? q1 * q1 : p1 * q1;
        acc = __builtin_amdgcn_wmma_f32_16x16x4_f32(false, a, false, wbf[kk >> 2], (short)0, acc, false, false);
      }
      // Unconditional stores: t1 rows 26..31 are dead padding (never read with
      // nonzero weight), so no EXEC-mask predication is needed.
#pragma unroll
      for (int r = 0; r < 8; ++r) t1[d][rowbase + r + 8 * half][lr] = acc[r];
    }
    __syncthreads();

    // ---- H blur as WMMA: OUT[h'][w] = sum_k hbm[h'][k] * t1[d][k][w].
    // k in [26,32) hits dead t1 rows, but hbm[:,k>=26] == 0.
#pragma unroll
    for (int i = 0; i < 2; ++i) {
      int d = wv + 8 * i;
      v8f acc = {};
#pragma unroll
      for (int kk = 0; kk < 32; kk += 4) {
        v2f b;
        b[0] = t1[d][kk + koff + 0][lr];
        b[1] = t1[d][kk + koff + 1][lr];
        acc = __builtin_amdgcn_wmma_f32_16x16x4_f32(false, hbf[kk >> 2], false, b, (short)0, acc, false, false);
      }
#pragma unroll
      for (int r = 0; r < 8; ++r) t2[d][(r + 8 * half) * 16 + lr] = acc[r];
    }
    __syncthreads();

    // ---- D blur as WMMA: OUT[d'][s] = sum_k g16[d'][k] * t2[k][s].
#pragma unroll
    for (int i = 0; i < 2; ++i) {
      int c = wv + 8 * i;
      v8f acc = {};
#pragma unroll
      for (int kk = 0; kk < 16; kk += 4) {
        v2f b;
        b[0] = t2[kk + koff + 0][c * 16 + lr];
        b[1] = t2[kk + koff + 1][c * 16 + lr];
        acc = __builtin_amdgcn_wmma_f32_16x16x4_f32(false, gf[kk >> 2], false, b, (short)0, acc, false, false);
      }
      blur[f][i] = acc;
    }
    __syncthreads();
  }

  // SSIM map in registers on the WMMA D layout; apply rerange analytically:
  // mu' = A*mu + B ; sigma'^2 = A^2 sigma^2 ; sigma12' = A1*A2*sigma12.
  float ssum = 0.f;
#pragma unroll
  for (int i = 0; i < 2; ++i) {
#pragma unroll
    for (int r = 0; r < 8; ++r) {
      float m1r = blur[0][i][r], m2r = blur[1][i][r];
      float bpp = blur[2][i][r], bgg = blur[3][i][r], bpg = blur[4][i][r];
      float mu1 = fmaf(m1r, ap, bp);
      float mu2 = fmaf(m2r, ag, bg);
      float s1  = ap * ap * (bpp - m1r * m1r);
      float s2  = ag * ag * (bgg - m2r * m2r);
      float s12 = ap * ag * (bpg - m1r * m2r);
      float mu1s = mu1 * mu1, mu2s = mu2 * mu2, mu12 = mu1 * mu2;
      float v1 = 2.f * s12 + C2F;
      float v2 = s1 + s2 + C2F;
      ssum += ((2.f * mu12 + C1F) * v1) / ((mu1s + mu2s + C1F) * v2);
    }
  }

  // Block reductions -> per-batch double accumulators.
  red[tid] = msum;
  __syncthreads();
  for (int s = 128; s > 0; s >>= 1) { if (tid < s) red[tid] += red[tid + s]; __syncthreads(); }
  float mtot = red[0];
  __syncthreads();
  red[tid] = ssum;
  __syncthreads();
  for (int s = 128; s > 0; s >>= 1) { if (tid < s) red[tid] += red[tid + s]; __syncthreads(); }
  if (tid == 0) {
    double* mseacc = (double*)wsp;
    double* ssacc  = (double*)wsp + 4;
    atomicAdd(&mseacc[n], (double)mtot);
    atomicAdd(&ssacc[n], (double)red[0]);
  }
}

__global__ void k_final(const void* wsp, float* out, int out_size) {
  if (threadIdx.x != 0 || blockIdx.x != 0) return;
  const double* mse = (const double*)wsp;
  const double* ss  = mse + 4;
  const double inv = 1.0 / (16.0 * 512.0 * 512.0);
  double psnr = 0.0, ssim = 0.0;
  for (int nidx = 0; nidx < 4; ++nidx) {
    double m = mse[nidx] * inv;
    psnr += 10.0 * log10(1.0 / m);
    ssim += ss[nidx] * inv;
  }
  if (out_size > 0) out[0] = (float)psnr;
  if (out_size > 1) out[1] = (float)ssim;
  if (out_size > 2) out[2] = 4.0f;
}

extern "C" void kernel_launch(void* const* d_in, const int* in_sizes, int n_in,
                              void* d_out, int out_size, void* d_ws, size_t ws_size,
                              hipStream_t stream) {
  const float* pred = (const float*)d_in[0];
  const float* gt   = (const float*)d_in[1];
  float* out = (float*)d_out;
  unsigned long long n = (unsigned long long)in_sizes[0];
  unsigned* mm = (unsigned*)((char*)d_ws + 64);

  k_init<<<1, 64, 0, stream>>>(d_ws);
  k_minmax<<<2048, 256, 0, stream>>>(pred, n, mm, 0);
  k_minmax<<<2048, 256, 0, stream>>>(gt,   n, mm, 2);
  k_scale<<<1, 64, 0, stream>>>(d_ws);
  dim3 grid(512 / 16, 512 / 16, 4);
  k_main<<<grid, 256, 0, stream>>>(pred, gt, d_ws);
  k_final<<<1, 64, 0, stream>>>(d_ws, out, out_size);
}